// RoIAlignRotated_2327872274502
// MI455X (gfx1250) — compile-verified
//
#include <hip/hip_runtime.h>
#include <math.h>

// RoIAlignRotated forward for MI455X (gfx1250, wave32).
// (1) transpose features NCHW->NHWC into d_ws so channel gathers are
//     contiguous (1KB per corner, full cachelines, L2-resident: 64MB << 192MB);
// (2) one 64-thread block per ROI, lane owns 4 channels (b128 gathers),
//     ROI row broadcast via gfx1250 async global->LDS load (ASYNCcnt path),
//     uniform geometry scalarized by the compiler onto the SALU.
// ph/pw loops kept rolled: hot loop fits in I$; latency hiding comes from the
// 16 independent b128 loads per bin plus many resident blocks per WGP.

#define POUT 7
#define SAMP 2
#define SSCALE 0.125f

static constexpr int Cc = 256;
static constexpr int Hh = 128;
static constexpr int Ww = 128;

// ---------------- Kernel 1: NCHW -> NHWC transpose (tiled via LDS) ----------
__global__ __launch_bounds__(256) void nchw_to_nhwc(const float* __restrict__ in,
                                                    float* __restrict__ out)
{
    __shared__ float tile[32][33];              // +1 pad: no bank conflicts
    const int b   = blockIdx.z;
    const int hw0 = blockIdx.x * 32;            // position within H*W
    const int c0  = blockIdx.y * 32;            // channel tile
    const int tx  = threadIdx.x;                // 0..31
    const int ty  = threadIdx.y;                // 0..7

    const float* src = in  + (size_t)b * Cc * Hh * Ww;
    float*       dst = out + (size_t)b * Hh * Ww * Cc;

#pragma unroll
    for (int i = 0; i < 32; i += 8)             // read coalesced along hw
        tile[ty + i][tx] = src[(size_t)(c0 + ty + i) * (Hh * Ww) + hw0 + tx];

    __syncthreads();

#pragma unroll
    for (int i = 0; i < 32; i += 8)             // write coalesced along c
        dst[(size_t)(hw0 + ty + i) * Cc + c0 + tx] = tile[tx][ty + i];
}

// ---------------- corner fetch: 4 consecutive channels ----------------------
__device__ __forceinline__ float4 fetch4(const float* __restrict__ f,
                                         int b, int y, int x, int c0, int nhwc)
{
    if (nhwc) {
        // base + ((b*H + y)*W + x)*C + c0 ; c0 % 4 == 0 -> 16B aligned
        const float* p = f + (((size_t)b * Hh + y) * Ww + x) * Cc + c0;
        return *(const float4*)p;
    } else {
        const float* p = f + (((size_t)(b * Cc + c0)) * Hh + y) * Ww + x;
        float4 r;
        r.x = p[0 * Hh * Ww];
        r.y = p[1 * Hh * Ww];
        r.z = p[2 * Hh * Ww];
        r.w = p[3 * Hh * Ww];
        return r;
    }
}

// ---------------- Kernel 2: main RoIAlignRotated ----------------------------
__global__ __launch_bounds__(64) void roi_align_rotated_fwd(
    const float* __restrict__ feat,   // NHWC (d_ws) or NCHW (fallback)
    const float* __restrict__ rois,   // (N, 6)
    float* __restrict__ out,          // (N, C, 7, 7)
    int nhwc)
{
    const int n  = blockIdx.x;
    const int t  = threadIdx.x;       // 0..63
    const int c0 = t * 4;             // 4 consecutive channels per lane

    __shared__ float sroi[8];

    // gfx1250 async global->LDS broadcast of this block's ROI row (6 floats).
    // Lanes 0..5 of wave 0 each copy one dword; ASYNCcnt tracks completion.
    if (t < 6) {
        const float* gp  = rois + (size_t)n * 6 + t;
        unsigned     lds = (unsigned)(size_t)(&sroi[t]);   // low 32 bits = LDS offset
        asm volatile("global_load_async_to_lds_b32 %0, %1, off"
                     :: "v"(lds), "v"(gp) : "memory");
    }
    asm volatile("s_wait_asynccnt 0x0" ::: "memory");
    __syncthreads();

    const int   b  = (int)sroi[0];
    const float cx = sroi[1] * SSCALE;
    const float cy = sroi[2] * SSCALE;
    const float rw = fmaxf(sroi[3] * SSCALE, 1.0f);
    const float rh = fmaxf(sroi[4] * SSCALE, 1.0f);
    const float th = sroi[5];
    const float ct = cosf(th);
    const float st = sinf(th);
    const float bh = rh * (1.0f / POUT);
    const float bw = rw * (1.0f / POUT);

    const size_t cstride = (size_t)POUT * POUT;                    // 49
    float* obase = out + (((size_t)n * Cc + c0) * cstride);

#pragma unroll 1
    for (int ph = 0; ph < POUT; ++ph) {
#pragma unroll 1
        for (int pw = 0; pw < POUT; ++pw) {
            float ax = 0.f, ay = 0.f, az = 0.f, aw = 0.f;
#pragma unroll
            for (int sy = 0; sy < SAMP; ++sy) {
                const float gy = (float)ph + ((float)sy + 0.5f) * (1.0f / SAMP);
                const float yy = -0.5f * rh + gy * bh;
#pragma unroll
                for (int sx = 0; sx < SAMP; ++sx) {
                    const float gx = (float)pw + ((float)sx + 0.5f) * (1.0f / SAMP);
                    const float xx = -0.5f * rw + gx * bw;

                    float x = xx * ct - yy * st + cx;
                    float y = xx * st + yy * ct + cy;

                    // uniform (scalar-unit) validity test, exactly as reference
                    if (!((y > -1.0f) & (y < (float)Hh) &
                          (x > -1.0f) & (x < (float)Ww)))
                        continue;                 // contributes exactly 0

                    y = fminf(fmaxf(y, 0.0f), (float)(Hh - 1));
                    x = fminf(fmaxf(x, 0.0f), (float)(Ww - 1));
                    const int y0 = (int)floorf(y);
                    const int x0 = (int)floorf(x);
                    const int y1 = min(y0 + 1, Hh - 1);
                    const int x1 = min(x0 + 1, Ww - 1);
                    const float ly = y - (float)y0;
                    const float lx = x - (float)x0;
                    const float hy = 1.0f - ly, hx = 1.0f - lx;
                    const float w00 = hy * hx, w01 = hy * lx;
                    const float w10 = ly * hx, w11 = ly * lx;

                    const float4 v00 = fetch4(feat, b, y0, x0, c0, nhwc);
                    const float4 v01 = fetch4(feat, b, y0, x1, c0, nhwc);
                    const float4 v10 = fetch4(feat, b, y1, x0, c0, nhwc);
                    const float4 v11 = fetch4(feat, b, y1, x1, c0, nhwc);

                    ax += v00.x * w00 + v01.x * w01 + v10.x * w10 + v11.x * w11;
                    ay += v00.y * w00 + v01.y * w01 + v10.y * w10 + v11.y * w11;
                    az += v00.z * w00 + v01.z * w01 + v10.z * w10 + v11.z * w11;
                    aw += v00.w * w00 + v01.w * w01 + v10.w * w10 + v11.w * w11;
                }
            }
            // Regular (RT) stores: output (51MB) fits in the 192MB L2, and each
            // 128B line is completed by temporally-spread 4B stores — keep lines
            // resident so L2 write-combines and writes back to HBM once.
            const float inv = 1.0f / (SAMP * SAMP);
            const size_t o = (size_t)ph * POUT + pw;
            obase[o]               = ax * inv;
            obase[o + cstride]     = ay * inv;
            obase[o + 2 * cstride] = az * inv;
            obase[o + 3 * cstride] = aw * inv;
        }
    }
}

// ---------------- launch -----------------------------------------------------
extern "C" void kernel_launch(void* const* d_in, const int* in_sizes, int n_in,
                              void* d_out, int out_size, void* d_ws, size_t ws_size,
                              hipStream_t stream)
{
    const float* feat = (const float*)d_in[0];   // (B, 256, 128, 128) fp32
    const float* rois = (const float*)d_in[1];   // (N, 6) fp32
    float*       out  = (float*)d_out;           // (N, 256, 7, 7) fp32

    const int nrois = in_sizes[1] / 6;
    const int B     = in_sizes[0] / (Cc * Hh * Ww);

    const size_t need = (size_t)in_sizes[0] * sizeof(float);   // NHWC copy
    const int nhwc = (ws_size >= need) ? 1 : 0;

    const float* f = feat;
    if (nhwc) {
        dim3 tg((Hh * Ww) / 32, Cc / 32, B);     // (512, 8, B)
        nchw_to_nhwc<<<tg, dim3(32, 8), 0, stream>>>(feat, (float*)d_ws);
        f = (const float*)d_ws;
    }

    roi_align_rotated_fwd<<<dim3(nrois), dim3(64), 0, stream>>>(f, rois, out, nhwc);
}